// SinkhornKnoppSolver_27530740367399
// MI455X (gfx1250) — compile-verified
//
#include <hip/hip_runtime.h>
#include <math.h>

// Problem constants
#define BB   16
#define NN   1024
#define MM   1024
#define DD   64
#define EPSI 0.01f
#define INV_EPS 100.0f
#define ITERS 100
#define THRESH 1e-4f
// log(1/1024 + 1e-8)
#define LOGA (-6.93146157e0f)

typedef float v8f __attribute__((ext_vector_type(8)));
typedef float v2f __attribute__((ext_vector_type(2)));

// ---------------- workspace layout (in floats) ----------------
// C      : [BB*NN*MM]            offset 0
// xn     : [BB*NN]               offset CBASE + 0*SL
// yn     : [BB*MM]               offset CBASE + 1*SL
// u      : [BB*NN]               offset CBASE + 2*SL
// v      : [BB*MM]               offset CBASE + 3*SL
// du     : [BB*NN]               offset CBASE + 4*SL
// dv     : [BB*MM]               offset CBASE + 5*SL
// part   : [BB*NN]               offset CBASE + 6*SL
// done   : int                   offset CBASE + 7*SL
#define CBASE ((size_t)BB * NN * MM)
#define SL    ((size_t)BB * NN)

// ---------------------------------------------------------------
// Init: zero u, v, done; compute squared norms xn, yn.
__global__ void sk_init(const float* __restrict__ x, const float* __restrict__ y,
                        float* __restrict__ xn, float* __restrict__ yn,
                        float* __restrict__ u, float* __restrict__ v,
                        int* __restrict__ done) {
  int tid = blockIdx.x * blockDim.x + threadIdx.x;
  if (tid == 0) *done = 0;
  if (tid < BB * NN) {
    const float* r = x + (size_t)tid * DD;
    float s = 0.f;
    #pragma unroll 8
    for (int k = 0; k < DD; ++k) s += r[k] * r[k];
    xn[tid] = s;
    u[tid] = 0.f;
  } else if (tid < 2 * BB * NN) {
    int t = tid - BB * NN;
    const float* r = y + (size_t)t * DD;
    float s = 0.f;
    #pragma unroll 8
    for (int k = 0; k < DD; ++k) s += r[k] * r[k];
    yn[t] = s;
    v[t] = 0.f;
  }
}

// ---------------------------------------------------------------
// WMMA cost-matrix build: C[b,i,j] = xn[i] + yn[j] - 2 * x_i . y_j
// One wave per 16x16 output tile; K-loop of 16 x V_WMMA_F32_16X16X4_F32.
__global__ void sk_costmat_wmma(const float* __restrict__ x, const float* __restrict__ y,
                                const float* __restrict__ xn, const float* __restrict__ yn,
                                float* __restrict__ C) {
  const int wave = threadIdx.x >> 5;
  const int lane = threadIdx.x & 31;
  const int tile = blockIdx.x * 8 + wave;          // 0 .. BB*64*64-1
  const int b = tile >> 12;                        // / (64*64)
  const int rt = (tile >> 6) & 63;                 // i-tile
  const int ct = tile & 63;                        // j-tile

  const float* Xb = x + (size_t)b * NN * DD;
  const float* Yb = y + (size_t)b * MM * DD;

  const int m  = lane & 15;        // M (for A) / N (for B)
  const int hi = lane >> 4;
  const int kb = hi * 2;           // K sub-offset per ISA 16x4 f32 layout

  const float* xrow = Xb + (size_t)(rt * 16 + m) * DD;  // A row i0+m
  const float* yrow = Yb + (size_t)(ct * 16 + m) * DD;  // B col j0+n (n=m)

  v8f acc = {};
  #pragma unroll
  for (int k0 = 0; k0 < DD; k0 += 4) {
    v2f a = *(const v2f*)(xrow + k0 + kb);
    v2f bm = *(const v2f*)(yrow + k0 + kb);
    acc = __builtin_amdgcn_wmma_f32_16x16x4_f32(
        /*neg_a=*/false, a, /*neg_b=*/false, bm,
        /*c_mod=*/(short)0, acc, /*reuse_a=*/false, /*reuse_b=*/false);
  }

  // C/D layout: VGPR r -> rows (r + 8*hi), cols lane&15
  const int i0 = rt * 16 + hi * 8;
  const int j  = ct * 16 + (lane & 15);
  const float ynj = yn[b * MM + j];
  float* Cout = C + ((size_t)b * NN + i0) * MM + j;
  #pragma unroll
  for (int r = 0; r < 8; ++r) {
    float xni = xn[b * NN + i0 + r];
    Cout[(size_t)r * MM] = xni + ynj - 2.0f * acc[r];
  }
}

__device__ __forceinline__ void lse_merge(float& m, float& s, float m2, float s2) {
  float M_ = fmaxf(m, m2);
  s = s * __expf(m - M_) + s2 * __expf(m2 - M_);
  m = M_;
}

// ---------------------------------------------------------------
// Row update: one wave per row i.  u[b,i] updated in place; du = |u_new - u|.
__global__ void sk_row_update(const float* __restrict__ C,
                              float* __restrict__ u, const float* __restrict__ v,
                              float* __restrict__ du, const int* __restrict__ done) {
  if (*done) return;
  const int wave = threadIdx.x >> 5;
  const int lane = threadIdx.x & 31;
  const int row = blockIdx.x * 8 + wave;   // b*NN + i
  const int b = row >> 10;
  const float ui = u[row];
  const float* Crow = C + (size_t)row * MM;
  const float* vb = v + b * MM;

  float m = -3.0e38f, s = 0.f;
  #pragma unroll 4
  for (int j = lane; j < MM; j += 32) {
    float sij = (ui + vb[j] - Crow[j]) * INV_EPS;
    if (sij > m) { s = s * __expf(m - sij) + 1.0f; m = sij; }
    else         { s += __expf(sij - m); }
  }
  // wave32 cross-lane tree merge
  #pragma unroll
  for (int off = 16; off >= 1; off >>= 1) {
    float m2 = __shfl_xor(m, off, 32);
    float s2 = __shfl_xor(s, off, 32);
    lse_merge(m, s, m2, s2);
  }
  if (lane == 0) {
    float lse = m + __logf(s);
    float unew = EPSI * (LOGA - lse) + ui;
    du[row] = fabsf(unew - ui);
    u[row] = unew;
  }
}

// ---------------------------------------------------------------
// Column update: workgroup handles 32 columns; 8 waves stride rows (coalesced),
// cross-wave merge in LDS.  v updated in place; dv = |v_new - v|.
__global__ void sk_col_update(const float* __restrict__ C,
                              const float* __restrict__ u, float* __restrict__ v,
                              float* __restrict__ dv, const int* __restrict__ done) {
  if (*done) return;
  const int wave = threadIdx.x >> 5;
  const int lane = threadIdx.x & 31;
  const int b = blockIdx.x >> 5;          // 32 column-tiles per batch
  const int j = (blockIdx.x & 31) * 32 + lane;
  const float vj = v[b * MM + j];
  const float* Cb = C + (size_t)b * NN * MM;
  const float* ub = u + b * NN;

  float m = -3.0e38f, s = 0.f;
  for (int i = wave; i < NN; i += 8) {
    float sij = (ub[i] + vj - Cb[(size_t)i * MM + j]) * INV_EPS;
    if (sij > m) { s = s * __expf(m - sij) + 1.0f; m = sij; }
    else         { s += __expf(sij - m); }
  }

  __shared__ float sm[8][32];
  __shared__ float ss[8][32];
  sm[wave][lane] = m;
  ss[wave][lane] = s;
  __syncthreads();

  if (wave == 0) {
    #pragma unroll
    for (int w = 1; w < 8; ++w) lse_merge(m, s, sm[w][lane], ss[w][lane]);
    float lse = m + __logf(s);
    float vnew = EPSI * (LOGA - lse) + vj;
    dv[b * MM + j] = fabsf(vnew - vj);
    v[b * MM + j] = vnew;
  }
}

// ---------------------------------------------------------------
// Finalize one iteration: diff = mean_b( sum|du| + sum|dv| ); set done flag.
// Fixed-partition tree reduction -> bit-deterministic across replays.
__global__ void sk_finalize(const float* __restrict__ du, const float* __restrict__ dv,
                            int* __restrict__ done) {
  if (*done) return;
  __shared__ float red[256];
  float acc = 0.f;
  for (int idx = threadIdx.x; idx < BB * NN; idx += 256) acc += du[idx] + dv[idx];
  red[threadIdx.x] = acc;
  __syncthreads();
  for (int off = 128; off >= 1; off >>= 1) {
    if (threadIdx.x < off) red[threadIdx.x] += red[threadIdx.x + off];
    __syncthreads();
  }
  if (threadIdx.x == 0) {
    float diff = red[0] / (float)BB;   // mean over batches of per-batch sums
    if (diff < THRESH) *done = 1;
  }
}

// ---------------------------------------------------------------
// Final cost, stage 1: per-row partial of sum_ij exp(kernel)*C.
__global__ void sk_cost_part(const float* __restrict__ C,
                             const float* __restrict__ u, const float* __restrict__ v,
                             float* __restrict__ part) {
  const int wave = threadIdx.x >> 5;
  const int lane = threadIdx.x & 31;
  const int row = blockIdx.x * 8 + wave;
  const int b = row >> 10;
  const float ui = u[row];
  const float* Crow = C + (size_t)row * MM;
  const float* vb = v + b * MM;
  float acc = 0.f;
  #pragma unroll 4
  for (int j = lane; j < MM; j += 32) {
    float c = Crow[j];
    acc += __expf((ui + vb[j] - c) * INV_EPS) * c;
  }
  #pragma unroll
  for (int off = 16; off >= 1; off >>= 1) acc += __shfl_xor(acc, off, 32);
  if (lane == 0) part[row] = acc;
}

// Final cost, stage 2: reduce per-batch partials (deterministic tree).
__global__ void sk_cost_final(const float* __restrict__ part, float* __restrict__ out) {
  const int b = blockIdx.x;
  __shared__ float red[256];
  float acc = 0.f;
  for (int i = threadIdx.x; i < NN; i += 256) acc += part[b * NN + i];
  red[threadIdx.x] = acc;
  __syncthreads();
  for (int off = 128; off >= 1; off >>= 1) {
    if (threadIdx.x < off) red[threadIdx.x] += red[threadIdx.x + off];
    __syncthreads();
  }
  if (threadIdx.x == 0) out[b] = red[0];
}

// ---------------------------------------------------------------
extern "C" void kernel_launch(void* const* d_in, const int* in_sizes, int n_in,
                              void* d_out, int out_size, void* d_ws, size_t ws_size,
                              hipStream_t stream) {
  const float* x = (const float*)d_in[0];
  const float* y = (const float*)d_in[1];
  float* out = (float*)d_out;

  float* ws = (float*)d_ws;
  float* C   = ws;
  float* xn  = ws + CBASE + 0 * SL;
  float* yn  = ws + CBASE + 1 * SL;
  float* u   = ws + CBASE + 2 * SL;
  float* v   = ws + CBASE + 3 * SL;
  float* du  = ws + CBASE + 4 * SL;
  float* dv  = ws + CBASE + 5 * SL;
  float* prt = ws + CBASE + 6 * SL;
  int*   done = (int*)(ws + CBASE + 7 * SL);

  (void)in_sizes; (void)n_in; (void)out_size; (void)ws_size;

  // norms + zero state + clear done flag (runs every call -> replay-safe)
  sk_init<<<(2 * BB * NN + 255) / 256, 256, 0, stream>>>(x, y, xn, yn, u, v, done);

  // build C with f32 WMMA: BB*64*64 tiles, 8 waves (tiles) per 256-thread block
  sk_costmat_wmma<<<(BB * 64 * 64) / 8, 256, 0, stream>>>(x, y, xn, yn, C);

  // Sinkhorn iterations (C stays L2-resident: 64MB < 192MB L2)
  for (int t = 0; t < ITERS; ++t) {
    sk_row_update<<<(BB * NN) / 8, 256, 0, stream>>>(C, u, v, du, done);
    sk_col_update<<<BB * (MM / 32), 256, 0, stream>>>(C, u, v, dv, done);
    sk_finalize<<<1, 256, 0, stream>>>(du, dv, done);
  }

  // transport cost
  sk_cost_part<<<(BB * NN) / 8, 256, 0, stream>>>(C, u, v, prt);
  sk_cost_final<<<BB, 256, 0, stream>>>(prt, out);
}